// ObjectTransModule_38920993636792
// MI455X (gfx1250) — compile-verified
//
#include <hip/hip_runtime.h>
#include <math.h>

// ---------------------------------------------------------------------------
// MI455X (gfx1250) implementation.
//   * Dense projections (lin1, Wih pre-gates, lin2) = large parallel GEMMs
//     (M = B*T = 16384) on V_WMMA_F32_16X16X4_F32, zero predication in the
//     K loop (odd shapes pre-padded in workspace).
//   * Recurrence: 1 workgroup (16 waves) per (chain, direction); Whh lives
//     entirely in VGPRs (128 regs/lane); per step 64 f32 WMMA + 1024 gate
//     lanes. The next timestep's pre-gate tile is double-buffered into LDS
//     with GLOBAL_LOAD_ASYNC_TO_LDS_B128 (ASYNCcnt), overlapping all global
//     latency with compute — nothing on the 2048-step serial chain touches
//     global memory synchronously.
// ---------------------------------------------------------------------------

typedef __attribute__((ext_vector_type(2))) float v2f;
typedef __attribute__((ext_vector_type(8))) float v8f;

#define BATCH 8
#define TLEN  2048
#define BT    (BATCH * TLEN)
#define HD    128

// ---------------------------- helpers --------------------------------------

__device__ __forceinline__ float sigf(float x) { return 1.0f / (1.0f + expf(-x)); }

__device__ __forceinline__ unsigned rotl32(unsigned v, int n) {
  return (v << n) | (v >> (32 - n));
}

// JAX threefry2x32, key = PRNGKey(42) -> (0, 42)
__device__ __forceinline__ void threefry2x32(unsigned k0, unsigned k1,
                                             unsigned& x0, unsigned& x1) {
  const unsigned ks2 = 0x1BD11BDAu ^ k0 ^ k1;
  const int R0[4] = {13, 15, 26, 6};
  const int R1[4] = {17, 29, 16, 24};
  x0 += k0; x1 += k1;
#pragma unroll
  for (int i = 0; i < 4; ++i) { x0 += x1; x1 = rotl32(x1, R0[i]); x1 ^= x0; }
  x0 += k1; x1 += ks2 + 1u;
#pragma unroll
  for (int i = 0; i < 4; ++i) { x0 += x1; x1 = rotl32(x1, R1[i]); x1 ^= x0; }
  x0 += ks2; x1 += k0 + 2u;
#pragma unroll
  for (int i = 0; i < 4; ++i) { x0 += x1; x1 = rotl32(x1, R0[i]); x1 ^= x0; }
  x0 += k0; x1 += k1 + 3u;
#pragma unroll
  for (int i = 0; i < 4; ++i) { x0 += x1; x1 = rotl32(x1, R1[i]); x1 ^= x0; }
  x0 += k1; x1 += ks2 + 4u;
#pragma unroll
  for (int i = 0; i < 4; ++i) { x0 += x1; x1 = rotl32(x1, R0[i]); x1 ^= x0; }
  x0 += ks2; x1 += k0 + 5u;
}

__device__ __forceinline__ float erfinv_f(float x) {
  float w = -logf((1.0f - x) * (1.0f + x));
  float p;
  if (w < 5.0f) {
    w -= 2.5f;
    p = 2.81022636e-08f;
    p = fmaf(p, w, 3.43273939e-07f);
    p = fmaf(p, w, -3.5233877e-06f);
    p = fmaf(p, w, -4.39150654e-06f);
    p = fmaf(p, w, 0.00021858087f);
    p = fmaf(p, w, -0.00125372503f);
    p = fmaf(p, w, -0.00417768164f);
    p = fmaf(p, w, 0.246640727f);
    p = fmaf(p, w, 1.50140941f);
  } else {
    w = sqrtf(w) - 3.0f;
    p = -0.000200214257f;
    p = fmaf(p, w, 0.000100950558f);
    p = fmaf(p, w, 0.00134934322f);
    p = fmaf(p, w, -0.00367342844f);
    p = fmaf(p, w, 0.00573950773f);
    p = fmaf(p, w, -0.0076224613f);
    p = fmaf(p, w, 0.00943887047f);
    p = fmaf(p, w, 1.00167406f);
    p = fmaf(p, w, 2.83297682f);
  }
  return p * x;
}

// jax.random.normal(key(42), (8,2048,3)) element e
__device__ __forceinline__ float jax_normal_noise(unsigned e) {
  unsigned half = (e >= 24576u) ? 1u : 0u;
  unsigned i = half ? (e - 24576u) : e;
  unsigned x0 = i, x1 = i + 24576u;
  threefry2x32(0u, 42u, x0, x1);
  unsigned bits = half ? x1 : x0;
  float f01 = __uint_as_float(0x3f800000u | (bits >> 9)) - 1.0f;
  float u = fmaxf(-0.99999994f, fmaf(f01, 1.99999994f, -0.99999994f));
  return 1.41421356237f * erfinv_f(u);
}

// ----------------------- feature / noise kernel ----------------------------

__global__ void feat_kernel(const float* __restrict__ hands,
                            const float* __restrict__ rot,
                            const float* __restrict__ trans,
                            float* __restrict__ lfeat,
                            float* __restrict__ rfeat,
                            float* __restrict__ blenl,
                            float* __restrict__ blenr) {
  int e = blockIdx.x * blockDim.x + threadIdx.x;
  if (e >= BT) return;
  float lp[3], rp[3], nz[3];
#pragma unroll
  for (int c = 0; c < 3; ++c) {
    lp[c] = hands[e * 6 + c];
    rp[c] = hands[e * 6 + 3 + c];
    nz[c] = trans[e * 3 + c] + 0.05f * jax_normal_noise((unsigned)(e * 3 + c));
  }
  // rot6d = R[:, :2] reshaped row-major: R00,R01,R10,R11,R20,R21
  const float* R = rot + (size_t)e * 9;
  float r6[6] = {R[0], R[1], R[3], R[4], R[6], R[7]};
  float* lf = lfeat + (size_t)e * 16;
  float* rf = rfeat + (size_t)e * 16;
#pragma unroll
  for (int c = 0; c < 6; ++c) { lf[c] = r6[c]; rf[c] = r6[c]; }
  float dl2 = 0.f, dr2 = 0.f;
#pragma unroll
  for (int c = 0; c < 3; ++c) {
    lf[6 + c] = lp[c];  rf[6 + c] = rp[c];
    lf[9 + c] = nz[c];  rf[9 + c] = nz[c];
    float dl = nz[c] - lp[c], dr = nz[c] - rp[c];
    lf[12 + c] = dl;    rf[12 + c] = dr;
    dl2 += dl * dl;     dr2 += dr * dr;
  }
  lf[15] = 0.f; rf[15] = 0.f;
  blenl[e] = sqrtf(dl2);
  blenr[e] = sqrtf(dr2);
}

// ----------------------- initial-state projection --------------------------

__global__ void proj_kernel(const float* __restrict__ init64,
                            const float* __restrict__ pw,   // (1024, 64)
                            const float* __restrict__ pb,   // (1024)
                            float* __restrict__ sout) {     // (8, 1024)
  int idx = blockIdx.x * blockDim.x + threadIdx.x;
  if (idx >= 8 * 1024) return;
  int b = idx >> 10, j = idx & 1023;
  float acc = pb[j];
#pragma unroll 8
  for (int k = 0; k < 64; ++k) acc += init64[b * 64 + k] * pw[j * 64 + k];
  sout[idx] = acc;
}

// ----------------------- weight padding kernel -----------------------------
// lin1_w (128,15) -> (128,16); lin2_w (3,256) -> (64,256); lin2_b (3) -> (64)

__global__ void pack_weights_kernel(const float* __restrict__ lin1_w,
                                    const float* __restrict__ lin2_w,
                                    const float* __restrict__ lin2_b,
                                    float* __restrict__ lin1p,
                                    float* __restrict__ lin2p,
                                    float* __restrict__ lin2bp) {
  int i = blockIdx.x * blockDim.x + threadIdx.x;
  if (i < 128 * 16) {
    int r = i >> 4, c = i & 15;
    lin1p[i] = (c < 15) ? lin1_w[r * 15 + c] : 0.f;
  }
  if (i < 64 * 256) {
    int r = i >> 8, c = i & 255;
    lin2p[i] = (r < 3) ? lin2_w[r * 256 + c] : 0.f;
  }
  if (i < 64) lin2bp[i] = (i < 3) ? lin2_b[i] : 0.f;
}

// --------------------------- WMMA GEMM -------------------------------------
// C[M,N](ldc,+coloff) = act(A[M,K](lda) @ W[N,K]^T(ldw) + b1 (+ b2))
// Requirements: M%64==0, N%64==0, K%4==0 (caller pre-pads). No predication
// in the K loop. Block = 256 threads = 8 waves = 4 M-tiles x 2 N-halves;
// each wave computes two 16x16 tiles sharing one A fragment.

__global__ __launch_bounds__(256)
void wmma_gemm_kernel(const float* __restrict__ A, int lda,
                      const float* __restrict__ W, int ldw,
                      const float* __restrict__ b1,
                      const float* __restrict__ b2,
                      float* __restrict__ C, int ldc, int coloff,
                      int M, int N, int K, int Nvalid, int act) {
  const int wave = threadIdx.x >> 5;
  const int lane = threadIdx.x & 31;
  const int lc16 = lane & 15, lhi = lane >> 4;
  const int mtile = blockIdx.y * 64 + (wave >> 1) * 16;
  const int nt0 = blockIdx.x * 64 + (wave & 1) * 32;
  const int nt1 = nt0 + 16;

  const float* Arow = A + (size_t)(mtile + lc16) * lda + 2 * lhi;
  const float* W0 = W + (size_t)(nt0 + lc16) * ldw + 2 * lhi;
  const float* W1 = W + (size_t)(nt1 + lc16) * ldw + 2 * lhi;

  v8f acc0 = {}, acc1 = {};
#pragma unroll 2
  for (int kk = 0; kk < K; kk += 4) {
    v2f a  = *(const v2f*)(Arow + kk);
    v2f w0 = *(const v2f*)(W0 + kk);
    v2f w1 = *(const v2f*)(W1 + kk);
    acc0 = __builtin_amdgcn_wmma_f32_16x16x4_f32(false, a, false, w0,
                                                 (short)0, acc0, false, false);
    acc1 = __builtin_amdgcn_wmma_f32_16x16x4_f32(false, a, false, w1,
                                                 (short)0, acc1, false, false);
  }

  const int col0 = nt0 + lc16, col1 = nt1 + lc16;
  float bb0 = b1[col0], bb1 = b1[col1];
  if (b2) { bb0 += b2[col0]; bb1 += b2[col1]; }
#pragma unroll
  for (int v = 0; v < 8; ++v) {
    int m = mtile + v + 8 * lhi;
    float v0 = acc0[v] + bb0;
    float v1 = acc1[v] + bb1;
    if (act == 1) { v0 = fmaxf(v0, 0.f); v1 = fmaxf(v1, 0.f); }
    if (col0 < Nvalid) C[(size_t)m * ldc + col0 + coloff] = v0;
    if (col1 < Nvalid) C[(size_t)m * ldc + col1 + coloff] = v1;
  }
}

// --------------------------- LSTM recurrence -------------------------------
// blockIdx.x = direction. 512 threads = 16 waves; wave w owns gate columns
// [32w, 32w+32). Whh fragments in registers; pre[t] tiles double-buffered
// into LDS via GLOBAL_LOAD_ASYNC_TO_LDS_B128 (ASYNCcnt).
// LDS: pre 2x8x512 (32KB) + z 8x512 (16KB) + h 16x128 (8KB) + c 8x128 (4KB).

__global__ __launch_bounds__(512)
void lstm_recur_kernel(const float* __restrict__ pre,    // [2][BT][512]
                       const float* __restrict__ Whh_f,  // (512,128)
                       const float* __restrict__ Whh_b,  // (512,128)
                       const float* __restrict__ s,      // (8,1024)
                       int layer,
                       float* __restrict__ y) {          // [BT][256]
  __shared__ float sh_pre[2][BATCH][512];
  __shared__ float sh_z[BATCH][512];
  __shared__ float sh_h[16][HD];
  __shared__ float sh_c[BATCH][HD];

  const int dir = blockIdx.x;
  const float* Whh = dir ? Whh_b : Whh_f;
  const float* pred = pre + (size_t)dir * BT * 512;
  const int tid = threadIdx.x;
  const int ci = 2 * layer + dir;

  // Async-copy helper: this thread's 32-byte slice of pre[t] -> sh_pre[buf].
  const int arow = tid >> 6;              // batch row 0..7
  const int acol = (tid & 63) * 8;        // 8 floats per thread
  auto issue_pre = [&](int buf, int t) {
    const float* ga = pred + (size_t)(arow * TLEN + t) * 512 + acol;
    unsigned lds = (unsigned)(unsigned long long)&sh_pre[buf][arow][acol];
    asm volatile("global_load_async_to_lds_b128 %0, %1, off"
                 :: "v"(lds), "v"(ga) : "memory");
    asm volatile("global_load_async_to_lds_b128 %0, %1, off offset:16"
                 :: "v"(lds), "v"(ga) : "memory");
  };

  for (int e = tid; e < 16 * HD; e += 512) {
    int m = e >> 7, k = e & 127;
    sh_h[m][k] = (m < BATCH) ? s[m * 1024 + ci * HD + k] : 0.f;
    if (m < BATCH) sh_c[m][k] = s[m * 1024 + 512 + ci * HD + k];
  }
  issue_pre(0, dir ? TLEN - 1 : 0);
  asm volatile("s_wait_asynccnt 0" ::: "memory");
  __syncthreads();

  const int wave = tid >> 5, lane = tid & 31;
  const int lc16 = lane & 15, lhi = lane >> 4;
  const int n0 = wave * 32, n1 = n0 + 16;
  const int col0 = n0 + lc16, col1 = n1 + lc16;

  // Preload this wave's Whh fragments into registers (loop-invariant).
  v2f wr0[32], wr1[32];
  {
    const float* W0 = Whh + (size_t)col0 * HD + 2 * lhi;
    const float* W1 = Whh + (size_t)col1 * HD + 2 * lhi;
#pragma unroll
    for (int kk = 0; kk < 32; ++kk) {
      wr0[kk] = *(const v2f*)(W0 + kk * 4);
      wr1[kk] = *(const v2f*)(W1 + kk * 4);
    }
  }

  for (int step = 0; step < TLEN; ++step) {
    const int t = dir ? (TLEN - 1 - step) : step;
    const int buf = step & 1;

    // Kick off async copy of the NEXT timestep's pre tile (overlaps GEMM).
    if (step + 1 < TLEN) issue_pre(buf ^ 1, dir ? t - 1 : t + 1);

    // hWhh = h @ Whh^T  (accumulators start at zero; pre added at store).
    v8f acc0 = {}, acc1 = {};
#pragma unroll
    for (int kk = 0; kk < 32; ++kk) {
      v2f a = *(const v2f*)&sh_h[lc16][kk * 4 + 2 * lhi];
      acc0 = __builtin_amdgcn_wmma_f32_16x16x4_f32(false, a, false, wr0[kk],
                                                   (short)0, acc0, false, false);
      acc1 = __builtin_amdgcn_wmma_f32_16x16x4_f32(false, a, false, wr1[kk],
                                                   (short)0, acc1, false, false);
    }
    // Rows 0..7 of D live in lanes 0..15; fold in pre[t] and store z.
    if (lhi == 0) {
#pragma unroll
      for (int v = 0; v < 8; ++v) {
        sh_z[v][col0] = acc0[v] + sh_pre[buf][v][col0];
        sh_z[v][col1] = acc1[v] + sh_pre[buf][v][col1];
      }
    }
    __syncthreads();

    // gates: i,f,g,o = z split along 512; update c,h
#pragma unroll
    for (int r = 0; r < 2; ++r) {
      int e = tid + r * 512;
      int b = e >> 7, k = e & 127;
      float zi = sh_z[b][k];
      float zf = sh_z[b][k + 128];
      float zg = sh_z[b][k + 256];
      float zo = sh_z[b][k + 384];
      float c = sigf(zf) * sh_c[b][k] + sigf(zi) * tanhf(zg);
      float h = sigf(zo) * tanhf(c);
      sh_c[b][k] = c;
      sh_h[b][k] = h;
      y[(size_t)(b * TLEN + t) * 256 + dir * HD + k] = h;
    }
    asm volatile("s_wait_asynccnt 0" ::: "memory");
    __syncthreads();
  }
}

// --------------------------- trans kernel ----------------------------------

__global__ void trans_kernel(const float* __restrict__ hands,
                             const float* __restrict__ rot,
                             const float* __restrict__ blenl,
                             const float* __restrict__ blenr,
                             float* __restrict__ out) {
  int e = blockIdx.x * blockDim.x + threadIdx.x;
  if (e >= BT) return;
  const float* R = rot + (size_t)e * 9;
  float* o = out + (size_t)e * 15;
  float ld[3] = {o[0], o[1], o[2]};
  float rd[3] = {o[3], o[4], o[5]};
  float bl = blenl[e], br = blenr[e];
#pragma unroll
  for (int i = 0; i < 3; ++i) {
    float rl = R[i * 3 + 0] * ld[0] + R[i * 3 + 1] * ld[1] + R[i * 3 + 2] * ld[2];
    float rr = R[i * 3 + 0] * rd[0] + R[i * 3 + 1] * rd[1] + R[i * 3 + 2] * rd[2];
    o[6 + i] = hands[e * 6 + i] + rl * bl;
    o[9 + i] = hands[e * 6 + 3 + i] + rr * br;
  }
}

// ----------------------------- FK scan -------------------------------------

__global__ void fk_kernel(const float* __restrict__ prob,
                          const float* __restrict__ hands,
                          const float* __restrict__ rot,
                          const float* __restrict__ trans,
                          float* __restrict__ out) {
  int b = threadIdx.x;
  if (b >= BATCH) return;
  int mode = 0;
  bool seen = false, prev_lc = false, prev_rc = false;
  float dvec[3] = {0, 0, 0}, held[3] = {0, 0, 0};
  float dist = 0.f;
  for (int t = 0; t < TLEN; ++t) {
    size_t e = (size_t)b * TLEN + t;
    bool lc = (t > 0) && (prob[e * 3 + 0] > 0.5f);
    bool rc = (t > 0) && (prob[e * 3 + 1] > 0.5f);
    bool ls = (t > 0) && lc && !prev_lc;
    bool rs = (t > 0) && rc && !prev_rc;
    prev_lc = lc; prev_rc = rc;
    int new_c = ls ? 1 : (rs ? 2 : 0);
    bool has_c = (mode == 1) ? lc : rc;
    bool has_a = (mode == 2) ? lc : rc;
    bool sw = (new_c == 0) && !has_c && has_a && (mode != 0);
    bool close = (new_c == 0) && !sw && !has_c && (mode != 0);
    int new_mode = (new_c != 0) ? new_c
                                : (sw ? (mode == 1 ? 2 : 1) : (close ? 0 : mode));
    bool starting = (new_c != 0) || sw;
    const float* R = rot + e * 9;
    const float* lp = hands + e * 6;
    const float* rp = lp + 3;
    float hand[3], gt[3], iobj[3], v[3];
#pragma unroll
    for (int i = 0; i < 3; ++i) {
      hand[i] = (new_mode == 1) ? lp[i] : rp[i];
      gt[i] = trans[e * 3 + i];
      iobj[i] = seen ? held[i] : gt[i];
      v[i] = iobj[i] - hand[i];
    }
    if (starting) {
#pragma unroll
      for (int i = 0; i < 3; ++i)
        dvec[i] = R[0 * 3 + i] * v[0] + R[1 * 3 + i] * v[1] + R[2 * 3 + i] * v[2];
      dist = sqrtf(v[0] * v[0] + v[1] * v[1] + v[2] * v[2]);
      seen = true;
    }
    bool in_seg = (new_mode != 0);
#pragma unroll
    for (int i = 0; i < 3; ++i) {
      float pf = hand[i] +
                 (R[i * 3 + 0] * dvec[0] + R[i * 3 + 1] * dvec[1] +
                  R[i * 3 + 2] * dvec[2]) * dist;
      float pos = in_seg ? pf : (seen ? held[i] : gt[i]);
      if (in_seg) held[i] = pf;
      out[e * 15 + 12 + i] = pos;
    }
    mode = new_mode;
  }
}

// ----------------------------- launcher ------------------------------------
// Flattening assumption (setup_inputs insertion order, tuples expanded):
//   0 hands_pos_feat  1 pred_hand_contact_prob  2 obj_rot  3 obj_trans
//   4 compressed_initial_state
//   5..26  lhand: lin1_w, lin1_b, lin2_w, lin2_b, proj_w, proj_b,
//                 4 x (Wih, Whh, bih, bhh)  [cells: L0f, L0b, L1f, L1b]
//   27..48 rhand: same layout

extern "C" void kernel_launch(void* const* d_in, const int* in_sizes, int n_in,
                              void* d_out, int out_size, void* d_ws, size_t ws_size,
                              hipStream_t stream) {
  const float* hands  = (const float*)d_in[0];
  const float* prob   = (const float*)d_in[1];
  const float* rot    = (const float*)d_in[2];
  const float* trans  = (const float*)d_in[3];
  const float* init64 = (const float*)d_in[4];
  float* out = (float*)d_out;
  float* ws = (float*)d_ws;

  size_t off = 0;
  float* lfeat  = ws + off; off += (size_t)BT * 16;
  float* rfeat  = ws + off; off += (size_t)BT * 16;
  float* blenl  = ws + off; off += BT;
  float* blenr  = ws + off; off += BT;
  float* sbuf   = ws + off; off += 8 * 1024;            // per-hand reuse
  float* lin1p  = ws + off; off += 128 * 16;            // padded lin1_w
  float* lin2p  = ws + off; off += 64 * 256;            // padded lin2_w
  float* lin2bp = ws + off; off += 64;                  // padded lin2_b
  float* x      = ws + off; off += (size_t)BT * 128;    // lin1 output
  float* y0     = ws + off; off += (size_t)BT * 256;    // layer-0 output
  float* y1     = ws + off; off += (size_t)BT * 256;    // layer-1 output
  float* pre    = ws + off; off += (size_t)2 * BT * 512; // 2 dirs of pre-gates

  feat_kernel<<<(BT + 255) / 256, 256, 0, stream>>>(hands, rot, trans,
                                                    lfeat, rfeat, blenl, blenr);

  for (int hand = 0; hand < 2; ++hand) {
    const int pbase = 5 + hand * 22;
    const float* lin1_w = (const float*)d_in[pbase + 0];
    const float* lin1_b = (const float*)d_in[pbase + 1];
    const float* lin2_w = (const float*)d_in[pbase + 2];
    const float* lin2_b = (const float*)d_in[pbase + 3];
    const float* proj_w = (const float*)d_in[pbase + 4];
    const float* proj_b = (const float*)d_in[pbase + 5];
    auto Wih = [&](int c) { return (const float*)d_in[pbase + 6 + c * 4 + 0]; };
    auto Whh = [&](int c) { return (const float*)d_in[pbase + 6 + c * 4 + 1]; };
    auto bih = [&](int c) { return (const float*)d_in[pbase + 6 + c * 4 + 2]; };
    auto bhh = [&](int c) { return (const float*)d_in[pbase + 6 + c * 4 + 3]; };
    const float* feat = hand ? rfeat : lfeat;

    pack_weights_kernel<<<(64 * 256 + 255) / 256, 256, 0, stream>>>(
        lin1_w, lin2_w, lin2_b, lin1p, lin2p, lin2bp);
    proj_kernel<<<(8 * 1024 + 255) / 256, 256, 0, stream>>>(init64, proj_w,
                                                            proj_b, sbuf);
    // lin1: (BT,16 padded) -> (BT,128), ReLU
    {
      dim3 g(128 / 64, BT / 64);
      wmma_gemm_kernel<<<g, 256, 0, stream>>>(feat, 16, lin1p, 16, lin1_b,
                                              nullptr, x, 128, 0,
                                              BT, 128, 16, 128, 1);
    }
    // layer 0: pre-gates (K=128) then recurrence
    {
      dim3 g(512 / 64, BT / 64);
      for (int d = 0; d < 2; ++d)
        wmma_gemm_kernel<<<g, 256, 0, stream>>>(x, 128, Wih(d), 128, bih(d),
                                                bhh(d),
                                                pre + (size_t)d * BT * 512,
                                                512, 0, BT, 512, 128, 512, 0);
      lstm_recur_kernel<<<2, 512, 0, stream>>>(pre, Whh(0), Whh(1), sbuf, 0, y0);
    }
    // layer 1: pre-gates (K=256) then recurrence
    {
      dim3 g(512 / 64, BT / 64);
      for (int d = 0; d < 2; ++d)
        wmma_gemm_kernel<<<g, 256, 0, stream>>>(y0, 256, Wih(2 + d), 256,
                                                bih(2 + d), bhh(2 + d),
                                                pre + (size_t)d * BT * 512,
                                                512, 0, BT, 512, 256, 512, 0);
      lstm_recur_kernel<<<2, 512, 0, stream>>>(pre, Whh(2), Whh(3), sbuf, 1, y1);
    }
    // lin2: (BT,256) -> dir (BT,3), straight into out cols hand*3..
    {
      dim3 g(64 / 64, BT / 64);
      wmma_gemm_kernel<<<g, 256, 0, stream>>>(y1, 256, lin2p, 256, lin2bp,
                                              nullptr, out, 15, hand * 3,
                                              BT, 64, 256, 3, 0);
    }
  }

  trans_kernel<<<(BT + 255) / 256, 256, 0, stream>>>(hands, rot, blenl, blenr, out);
  fk_kernel<<<1, 32, 0, stream>>>(prob, hands, rot, trans, out);
}